// WeightedMseLoss_83743272337942
// MI455X (gfx1250) — compile-verified
//
#include <hip/hip_runtime.h>

typedef __attribute__((ext_vector_type(2))) float v2f;
typedef __attribute__((ext_vector_type(8))) float v8f;

#define BLOCK    256
#define WAVES    (BLOCK / 32)
#define GRID_CAP 2048

// Full wave32 reduction using one V_WMMA_F32_16X16X4_F32.
// A (16x4 f32, 2 VGPRs/lane): v0 = acc, v1 = 0.
//   A[m][0]=acc[m], A[m][1]=0, A[m][2]=acc[16+m], A[m][3]=0
// B (4x16) = all ones  =>  D[m][n] = acc[m] + acc[16+m]  (independent of n).
// C/D layout: lane l<16 VGPR r = D[r][l]; lane l>=16 VGPR r = D[8+r][l-16].
// => sum of the 8 D VGPRs in lanes 0-15  = sum(acc[0..7])  + sum(acc[16..23])
//    sum of the 8 D VGPRs in lanes 16-31 = sum(acc[8..15]) + sum(acc[24..31])
// One shfl_xor(16) then yields the full 32-lane sum in every lane.
// Deterministic summation order (same every call). EXEC is all-1s at call sites.
__device__ __forceinline__ float wave_reduce_wmma(float acc) {
    v2f a; a[0] = acc;  a[1] = 0.0f;
    v2f b; b[0] = 1.0f; b[1] = 1.0f;
    v8f c = {};
    c = __builtin_amdgcn_wmma_f32_16x16x4_f32(
        /*neg_a=*/false, a, /*neg_b=*/false, b,
        /*c_mod=*/(short)0, c, /*reuse_a=*/false, /*reuse_b=*/false);
    float s = ((c[0] + c[1]) + (c[2] + c[3])) + ((c[4] + c[5]) + (c[6] + c[7]));
    s += __shfl_xor(s, 16, 32);
    return s;
}

__device__ __forceinline__ float weighted_sq(float xv, float yv,
                                             const float* __restrict__ lds_table) {
    float d = xv - yv;
    unsigned idx = (unsigned)(int)rintf(yv * 100.0f);   // v_rndne_f32
    idx = idx > 100u ? 100u : idx;
    return d * d * lds_table[idx];                      // ds_load_b32 gather
}

__global__ void __launch_bounds__(BLOCK)
wmse_partial_kernel(const float* __restrict__ x, const float* __restrict__ y,
                    const float* __restrict__ table,
                    float* __restrict__ partials, int n) {
    __shared__ float lds_table[104];
    __shared__ float lds_wave[WAVES];

    // Preload the 101-entry weight table into LDS (pad to 104).
    for (int i = threadIdx.x; i < 104; i += BLOCK)
        lds_table[i] = (i < 101) ? table[i] : 0.0f;
    __syncthreads();

    const float4* __restrict__ x4 = (const float4*)x;
    const float4* __restrict__ y4 = (const float4*)y;
    const int n4     = n >> 2;
    const int tid    = (int)(blockIdx.x * BLOCK + threadIdx.x);
    const int stride = (int)(gridDim.x * BLOCK);

    float acc = 0.0f;
    for (int i = tid; i < n4; i += stride) {
        // gfx1250 global_prefetch_b8 of next tile (speculative; OOB dropped).
        __builtin_prefetch(&x4[i + stride], 0, 1);
        __builtin_prefetch(&y4[i + stride], 0, 1);
        float4 xv = x4[i];          // global_load_b128
        float4 yv = y4[i];          // global_load_b128
        acc += weighted_sq(xv.x, yv.x, lds_table);
        acc += weighted_sq(xv.y, yv.y, lds_table);
        acc += weighted_sq(xv.z, yv.z, lds_table);
        acc += weighted_sq(xv.w, yv.w, lds_table);
    }

    // Scalar tail (n not divisible by 4) handled by block 0 only.
    if (blockIdx.x == 0) {
        for (int i = (n4 << 2) + (int)threadIdx.x; i < n; i += BLOCK)
            acc += weighted_sq(x[i], y[i], lds_table);
    }

    // Wave32 reduce via WMMA, then cross-wave via LDS.
    float ws = wave_reduce_wmma(acc);
    const int wave = (int)(threadIdx.x >> 5);
    if ((threadIdx.x & 31u) == 0u) lds_wave[wave] = ws;
    __syncthreads();
    if (threadIdx.x == 0) {
        float t = 0.0f;
#pragma unroll
        for (int i = 0; i < WAVES; ++i) t += lds_wave[i];
        partials[blockIdx.x] = t;
    }
}

__global__ void __launch_bounds__(BLOCK)
wmse_final_kernel(const float* __restrict__ partials, int num_partials,
                  float* __restrict__ out, float inv_n) {
    __shared__ float lds_wave[WAVES];
    float acc = 0.0f;
    for (int i = (int)threadIdx.x; i < num_partials; i += BLOCK)
        acc += partials[i];

    float ws = wave_reduce_wmma(acc);
    const int wave = (int)(threadIdx.x >> 5);
    if ((threadIdx.x & 31u) == 0u) lds_wave[wave] = ws;
    __syncthreads();
    if (threadIdx.x == 0) {
        float t = 0.0f;
#pragma unroll
        for (int i = 0; i < WAVES; ++i) t += lds_wave[i];
        out[0] = t * inv_n;
    }
}

extern "C" void kernel_launch(void* const* d_in, const int* in_sizes, int n_in,
                              void* d_out, int out_size, void* d_ws, size_t ws_size,
                              hipStream_t stream) {
    const float* x     = (const float*)d_in[0];
    const float* y     = (const float*)d_in[1];
    const float* table = (const float*)d_in[2];
    float* out      = (float*)d_out;
    float* partials = (float*)d_ws;     // GRID_CAP floats = 8 KB scratch

    const int n  = in_sizes[0];
    const int n4 = n >> 2;
    int blocks = (n4 + BLOCK - 1) / BLOCK;
    if (blocks > GRID_CAP) blocks = GRID_CAP;
    if (blocks < 1) blocks = 1;

    wmse_partial_kernel<<<blocks, BLOCK, 0, stream>>>(x, y, table, partials, n);
    wmse_final_kernel<<<1, BLOCK, 0, stream>>>(partials, blocks, out,
                                               1.0f / (float)n);
}